// DepthwiseConv1d_58746562674739
// MI455X (gfx1250) — compile-verified
//
#include <hip/hip_runtime.h>
#include <stdint.h>

#define B_    8
#define S_    4096
#define F_    2048
#define K_    4
#define CACHE (K_ - 1)
#define CHUNK 64
#define NG    (CHUNK / 4)
#define TPB   256

typedef float v4f __attribute__((ext_vector_type(4)));

// One output row-vector (4 features) from a 4-row window.
__device__ __forceinline__ v4f conv_step(v4f a, v4f b, v4f c, v4f d,
                                         v4f t0, v4f t1, v4f t2, v4f t3,
                                         v4f bs) {
    v4f o;
    o.x = fmaf(t0.x, a.x, fmaf(t0.y, b.x, fmaf(t0.z, c.x, fmaf(t0.w, d.x, bs.x))));
    o.y = fmaf(t1.x, a.y, fmaf(t1.y, b.y, fmaf(t1.z, c.y, fmaf(t1.w, d.y, bs.y))));
    o.z = fmaf(t2.x, a.z, fmaf(t2.y, b.z, fmaf(t2.z, c.z, fmaf(t2.w, d.z, bs.z))));
    o.w = fmaf(t3.x, a.w, fmaf(t3.y, b.w, fmaf(t3.z, c.w, fmaf(t3.w, d.w, bs.w))));
    return o;
}

__device__ __forceinline__ v4f nt_load4(const float* p) {
    return __builtin_nontemporal_load((const v4f*)p);
}
__device__ __forceinline__ void nt_store4(float* p, v4f v) {
    __builtin_nontemporal_store(v, (v4f*)p);
}

// Stage 4 rows (s..s+3, one b128 per lane per row) into smem buffer `buf`
// via async global->LDS DMA (GVS mode: SGPR64 base + VGPR32 byte offset).
// Tracked with ASYNCcnt.
#define STAGE(buf_, s_)                                                        \
    do {                                                                       \
        unsigned g0_ = (unsigned)(s_) * (F_ * 4u) + fbyte;                     \
        unsigned l0_ = lds_lane + (unsigned)(buf_) * (4u * TPB * 16u);         \
        asm volatile("global_load_async_to_lds_b128 %0, %1, %2"                \
                     :: "v"(l0_),                 "v"(g0_),                    \
                        "s"(xbase) : "memory");                                \
        asm volatile("global_load_async_to_lds_b128 %0, %1, %2"                \
                     :: "v"(l0_ + 1u * TPB * 16u), "v"(g0_ + 1u * F_ * 4u),    \
                        "s"(xbase) : "memory");                                \
        asm volatile("global_load_async_to_lds_b128 %0, %1, %2"                \
                     :: "v"(l0_ + 2u * TPB * 16u), "v"(g0_ + 2u * F_ * 4u),    \
                        "s"(xbase) : "memory");                                \
        asm volatile("global_load_async_to_lds_b128 %0, %1, %2"                \
                     :: "v"(l0_ + 3u * TPB * 16u), "v"(g0_ + 3u * F_ * 4u),    \
                        "s"(xbase) : "memory");                                \
    } while (0)

__global__ __launch_bounds__(TPB) void DepthwiseConv1d_kernel(
    const float* __restrict__ x,          // (B, S, F)
    const float* __restrict__ conv_state, // (B, F, CACHE)
    const float* __restrict__ weight,     // (F, 1, K)
    const float* __restrict__ bias,       // (F,)
    float* __restrict__ out,              // (B, S, F)
    float* __restrict__ new_state)        // (B, F, CACHE)
{
    __shared__ float smem[2][4][TPB][4];  // 32 KB: 2 bufs x 4 rows x lane x b128

    const int tid = threadIdx.x;
    const int f4  = blockIdx.x * TPB + tid;   // 0 .. F/4-1
    const int f0  = f4 * 4;
    const int s0  = blockIdx.y * CHUNK;
    const int b   = blockIdx.z;

    // Per-feature taps (float4 across K) and bias (float4 across features).
    v4f wr0 = *(const v4f*)(weight + (size_t)(f0 + 0) * K_);
    v4f wr1 = *(const v4f*)(weight + (size_t)(f0 + 1) * K_);
    v4f wr2 = *(const v4f*)(weight + (size_t)(f0 + 2) * K_);
    v4f wr3 = *(const v4f*)(weight + (size_t)(f0 + 3) * K_);
    v4f bs  = *(const v4f*)(bias + f0);

    const float* xb = x   + (size_t)b * S_ * F_ + f0;
    float*       ob = out + (size_t)b * S_ * F_ + f0;

    // Async staging parameters.
    const unsigned long long xbase =
        (unsigned long long)(uintptr_t)(x + (size_t)b * S_ * F_);  // SGPR64 base
    const unsigned fbyte    = (unsigned)f0 * 4u;                   // lane byte off
    const unsigned lds_lane =
        (unsigned)(uintptr_t)(void*)&smem[0][0][0][0] + (unsigned)tid * 16u;

    // Kick off group 0 fill while we set up the window.
    STAGE(0, s0);

    // Sliding window: padded rows s0-3, s0-2, s0-1 (components = 4 features).
    v4f w0, w1, w2;
    if (s0 == 0) {
        const float* cs = conv_state + ((size_t)b * F_ + f0) * CACHE;
        w0.x = cs[0 * CACHE + 0]; w0.y = cs[1 * CACHE + 0];
        w0.z = cs[2 * CACHE + 0]; w0.w = cs[3 * CACHE + 0];
        w1.x = cs[0 * CACHE + 1]; w1.y = cs[1 * CACHE + 1];
        w1.z = cs[2 * CACHE + 1]; w1.w = cs[3 * CACHE + 1];
        w2.x = cs[0 * CACHE + 2]; w2.y = cs[1 * CACHE + 2];
        w2.z = cs[2 * CACHE + 2]; w2.w = cs[3 * CACHE + 2];
    } else {
        w0 = nt_load4(xb + (size_t)(s0 - 3) * F_);
        w1 = nt_load4(xb + (size_t)(s0 - 2) * F_);
        w2 = nt_load4(xb + (size_t)(s0 - 1) * F_);
    }

    for (int g = 0; g < NG; ++g) {
        const int s   = s0 + 4 * g;
        const int buf = g & 1;

        if (g + 1 < NG) {
            STAGE(buf ^ 1, s + 4);
            // 8 async fills outstanding; <=4 left means group g has landed
            // (async loads complete in order).
            asm volatile("s_wait_asynccnt 4" ::: "memory");
        } else {
            asm volatile("s_wait_asynccnt 0" ::: "memory");
        }

        v4f r0 = *(const v4f*)&smem[buf][0][tid][0];
        v4f r1 = *(const v4f*)&smem[buf][1][tid][0];
        v4f r2 = *(const v4f*)&smem[buf][2][tid][0];
        v4f r3 = *(const v4f*)&smem[buf][3][tid][0];

        v4f o0 = conv_step(w0, w1, w2, r0, wr0, wr1, wr2, wr3, bs);
        v4f o1 = conv_step(w1, w2, r0, r1, wr0, wr1, wr2, wr3, bs);
        v4f o2 = conv_step(w2, r0, r1, r2, wr0, wr1, wr2, wr3, bs);
        v4f o3 = conv_step(r0, r1, r2, r3, wr0, wr1, wr2, wr3, bs);

        nt_store4(ob + (size_t)(s + 0) * F_, o0);
        nt_store4(ob + (size_t)(s + 1) * F_, o1);
        nt_store4(ob + (size_t)(s + 2) * F_, o2);
        nt_store4(ob + (size_t)(s + 3) * F_, o3);

        w0 = r1; w1 = r2; w2 = r3;  // window -> rows s+1, s+2, s+3
    }

    // After the last chunk the window holds x rows S-3, S-2, S-1:
    // exactly new_conv_state[b, f, 0..2].
    if (blockIdx.y == gridDim.y - 1) {
        float* ns = new_state + ((size_t)b * F_ + f0) * CACHE;
        ns[0 * CACHE + 0] = w0.x; ns[0 * CACHE + 1] = w1.x; ns[0 * CACHE + 2] = w2.x;
        ns[1 * CACHE + 0] = w0.y; ns[1 * CACHE + 1] = w1.y; ns[1 * CACHE + 2] = w2.y;
        ns[2 * CACHE + 0] = w0.z; ns[2 * CACHE + 1] = w1.z; ns[2 * CACHE + 2] = w2.z;
        ns[3 * CACHE + 0] = w0.w; ns[3 * CACHE + 1] = w1.w; ns[3 * CACHE + 2] = w2.w;
    }
}

extern "C" void kernel_launch(void* const* d_in, const int* in_sizes, int n_in,
                              void* d_out, int out_size, void* d_ws, size_t ws_size,
                              hipStream_t stream) {
    const float* x    = (const float*)d_in[0];
    const float* cs   = (const float*)d_in[1];
    const float* w    = (const float*)d_in[2];
    const float* bias = (const float*)d_in[3];

    float* out = (float*)d_out;
    float* ns  = out + (size_t)B_ * S_ * F_;  // tuple: (out, new_conv_state) flat

    dim3 grid(F_ / 4 / TPB, S_ / CHUNK, B_);  // (2, 64, 8)
    DepthwiseConv1d_kernel<<<grid, TPB, 0, stream>>>(x, cs, w, bias, out, ns);
}